// GNN1_17471926960715
// MI455X (gfx1250) — compile-verified
//
#include <hip/hip_runtime.h>

#define N_DRUG 846
#define D 64
#define K 1024
#define N_REL 200
#define RPAD 208            // 200 relations padded to 13*16 rows
#define BN_EPS 1e-5f

typedef __attribute__((ext_vector_type(2))) float v2f;
typedef __attribute__((ext_vector_type(8))) float v8f;

__device__ __forceinline__ float wave_reduce_add(float v) {
  #pragma unroll
  for (int off = 16; off > 0; off >>= 1) v += __shfl_xor(v, off, 32);
  return v;
}

// ---------------- kernel 0: b2sum[k] = sum_e b2[k,e] (drug-independent) ----
__global__ void __launch_bounds__(256) gnn_b2sum(const float* __restrict__ b2,
                                                 float* __restrict__ b2sum) {
  const int k = blockIdx.x * 256 + threadIdx.x;
  if (k < K) {
    const float* p = b2 + (size_t)k * D;
    float s = 0.f;
    #pragma unroll
    for (int e = 0; e < D; ++e) s += p[e];
    b2sum[k] = s;
  }
}

// ---------------- kernel 1: per-drug main kernel ---------------------------
// T = (drug_emb ⊙ rela_table) @ W1[n]   via fp32 WMMA 16x16x4
// score_k = softmax_k( Σ_d relu(T[r_k,d]+b1[k,d]) * w2sum[d] + b2sum[k] )
// drug_e[n] = [ Σ_k score_k * ent[tail_k] , drug_emb ]
__global__ void __launch_bounds__(256) gnn_main(
    const float* __restrict__ drug_table, const float* __restrict__ rela_table,
    const float* __restrict__ ent_table,  const float* __restrict__ W1g,
    const float* __restrict__ b1,         const float* __restrict__ W2g,
    const int*   __restrict__ drug_name,  const int* __restrict__ adj_tail,
    const int*   __restrict__ adj_rel,    const float* __restrict__ b2sum,
    float* __restrict__ drug_e)
{
  extern __shared__ float smem[];
  float* Atile = smem;                  // RPAD*64
  float* Ttile = Atile + RPAD * D;      // RPAD*64
  float* W1s   = Ttile + RPAD * D;      // 64*64
  float* demb  = W1s + D * D;           // 64
  float* w2sum = demb + D;              // 64
  float* score = w2sum + D;             // 1024
  float* red   = score + K;             // 512

  const int n    = blockIdx.x;
  const int t    = threadIdx.x;
  const int lane = t & 31;
  const int wave = t >> 5;

  const size_t wbase = (size_t)n * D * D;

  // drug embedding
  if (t < D) demb[t] = drug_table[(size_t)drug_name[n] * D + t];
  // stage W1[n] in LDS
  for (int i = t; i < D * D; i += 256) W1s[i] = W1g[wbase + i];
  // W2 row-sum partials: thread t covers row t/4, quarter t%4
  {
    const int d = t >> 2, q = t & 3;
    const float* p = W2g + wbase + d * D + q * 16;
    float s = 0.f;
    #pragma unroll
    for (int j = 0; j < 16; ++j) s += p[j];
    red[t] = s;
  }
  __syncthreads();
  if (t < D) w2sum[t] = red[4*t] + red[4*t+1] + red[4*t+2] + red[4*t+3];
  // A[r,d] = demb[d] * rela_table[r,d], zero-padded rows 200..207
  for (int i = t; i < RPAD * D; i += 256) {
    const int r = i >> 6, d = i & 63;
    Atile[i] = (r < N_REL) ? demb[d] * rela_table[r * D + d] : 0.f;
  }
  __syncthreads();

  // ---- WMMA: T = A @ W1   (13 M-tiles x 4 N-tiles, 16 k-steps of 4) ----
  const int half = lane >> 4;      // 0: lanes 0-15, 1: lanes 16-31
  const int l    = lane & 15;
  for (int tile = wave; tile < 13 * 4; tile += 8) {
    const int m0 = (tile % 13) * 16;
    const int n0 = (tile / 13) * 16;
    v8f c = {0.f, 0.f, 0.f, 0.f, 0.f, 0.f, 0.f, 0.f};
    const float* arow = Atile + (m0 + l) * D;
    #pragma unroll
    for (int kk = 0; kk < 16; ++kk) {
      const int k0 = kk * 4 + 2 * half;     // lane-half split on K
      v2f a, b;
      a[0] = arow[k0];
      a[1] = arow[k0 + 1];
      b[0] = W1s[k0 * D + n0 + l];
      b[1] = W1s[(k0 + 1) * D + n0 + l];
      c = __builtin_amdgcn_wmma_f32_16x16x4_f32(false, a, false, b,
                                                (short)0, c, false, false);
    }
    #pragma unroll
    for (int i = 0; i < 8; ++i)
      Ttile[(m0 + 8 * half + i) * D + n0 + l] = c[i];
  }
  __syncthreads();

  // ---- attention logits: wave w handles k = w*128 .. w*128+127 ----
  for (int i = 0; i < 128; ++i) {
    const int k = wave * 128 + i;
    const int r = adj_rel[(size_t)n * K + k];
    float v = fmaxf(Ttile[r * D + lane]      + b1[(size_t)k * D + lane],      0.f) * w2sum[lane]
            + fmaxf(Ttile[r * D + lane + 32] + b1[(size_t)k * D + lane + 32], 0.f) * w2sum[lane + 32];
    v = wave_reduce_add(v);
    if (lane == 0) score[k] = v + b2sum[k];
  }
  __syncthreads();

  // ---- softmax over K=1024 ----
  float lmax = -INFINITY;
  for (int k = t; k < K; k += 256) lmax = fmaxf(lmax, score[k]);
  red[t] = lmax;
  __syncthreads();
  for (int s = 128; s > 0; s >>= 1) {
    if (t < s) red[t] = fmaxf(red[t], red[t + s]);
    __syncthreads();
  }
  const float mx = red[0];
  __syncthreads();
  float lsum = 0.f;
  for (int k = t; k < K; k += 256) {
    const float e = __expf(score[k] - mx);
    score[k] = e;
    lsum += e;
  }
  red[t] = lsum;
  __syncthreads();
  for (int s = 128; s > 0; s >>= 1) {
    if (t < s) red[t] += red[t + s];
    __syncthreads();
  }
  const float inv = 1.f / red[0];
  __syncthreads();

  // ---- weighted entity gather ----
  float acc0 = 0.f, acc1 = 0.f;
  for (int i = 0; i < 128; ++i) {
    const int k = wave * 128 + i;
    const float sc = score[k] * inv;
    const size_t tl = (size_t)adj_tail[(size_t)n * K + k] * D;
    acc0 += sc * ent_table[tl + lane];
    acc1 += sc * ent_table[tl + lane + 32];
  }
  red[wave * 64 + lane]      = acc0;
  red[wave * 64 + lane + 32] = acc1;
  __syncthreads();
  if (t < D) {
    float s = 0.f;
    #pragma unroll
    for (int w = 0; w < 8; ++w) s += red[w * 64 + t];
    drug_e[(size_t)n * (2 * D) + t]     = s;        // weighted_ent
    drug_e[(size_t)n * (2 * D) + D + t] = demb[t];  // drug_emb
  }
}

// ---------------- kernel 2: y = relu(drug_e @ lin_w.T + lin_b) -------------
__global__ void __launch_bounds__(64) gnn_linear(
    const float* __restrict__ drug_e, const float* __restrict__ lin_w,
    const float* __restrict__ lin_b, float* __restrict__ y)
{
  __shared__ float de[2 * D];
  const int n = blockIdx.x, t = threadIdx.x;
  de[t]     = drug_e[(size_t)n * (2 * D) + t];
  de[t + D] = drug_e[(size_t)n * (2 * D) + D + t];
  __syncthreads();
  float acc = lin_b[t];
  const float* w = lin_w + (size_t)t * (2 * D);
  #pragma unroll 8
  for (int j = 0; j < 2 * D; ++j) acc += de[j] * w[j];
  y[(size_t)n * D + t] = fmaxf(acc, 0.f);
}

// ---------------- kernel 3: BatchNorm (training-mode, biased stats) --------
__global__ void __launch_bounds__(256) gnn_bn(
    const float* __restrict__ y, const float* __restrict__ bn_w,
    const float* __restrict__ bn_b, float* __restrict__ out)
{
  __shared__ float red[256];
  const int c = blockIdx.x, t = threadIdx.x;
  float s = 0.f;
  for (int r = t; r < N_DRUG; r += 256) s += y[(size_t)r * D + c];
  red[t] = s;
  __syncthreads();
  for (int st = 128; st > 0; st >>= 1) {
    if (t < st) red[t] += red[t + st];
    __syncthreads();
  }
  const float mean = red[0] / (float)N_DRUG;
  __syncthreads();
  float q = 0.f;
  for (int r = t; r < N_DRUG; r += 256) {
    const float d = y[(size_t)r * D + c] - mean;
    q += d * d;
  }
  red[t] = q;
  __syncthreads();
  for (int st = 128; st > 0; st >>= 1) {
    if (t < st) red[t] += red[t + st];
    __syncthreads();
  }
  const float var   = red[0] / (float)N_DRUG;
  const float scale = bn_w[c] * rsqrtf(var + BN_EPS);
  const float shift = bn_b[c];
  for (int r = t; r < N_DRUG; r += 256)
    out[(size_t)r * D + c] = (y[(size_t)r * D + c] - mean) * scale + shift;
}

// ---------------------------------------------------------------------------
extern "C" void kernel_launch(void* const* d_in, const int* in_sizes, int n_in,
                              void* d_out, int out_size, void* d_ws, size_t ws_size,
                              hipStream_t stream) {
  (void)in_sizes; (void)n_in; (void)out_size; (void)ws_size;
  const float* drug_table = (const float*)d_in[0];
  const float* rela_table = (const float*)d_in[1];
  const float* ent_table  = (const float*)d_in[2];
  const float* W1         = (const float*)d_in[3];
  const float* b1         = (const float*)d_in[4];
  const float* W2         = (const float*)d_in[5];
  const float* b2         = (const float*)d_in[6];
  const float* lin_w      = (const float*)d_in[7];
  const float* lin_b      = (const float*)d_in[8];
  const float* bn_w       = (const float*)d_in[9];
  const float* bn_b       = (const float*)d_in[10];
  const int*   drug_name  = (const int*)d_in[11];
  const int*   adj_tail   = (const int*)d_in[12];
  const int*   adj_rel    = (const int*)d_in[13];

  float* ws     = (float*)d_ws;
  float* b2sum  = ws;                       // 1024
  float* drug_e = ws + K;                   // 846*128
  float* y      = drug_e + N_DRUG * 2 * D;  // 846*64

  gnn_b2sum<<<dim3((K + 255) / 256), dim3(256), 0, stream>>>(b2, b2sum);

  const size_t lds = (size_t)(RPAD * D * 2 + D * D + D + D + K + 512) * sizeof(float);
  gnn_main<<<dim3(N_DRUG), dim3(256), lds, stream>>>(
      drug_table, rela_table, ent_table, W1, b1, W2,
      drug_name, adj_tail, adj_rel, b2sum, drug_e);

  gnn_linear<<<dim3(N_DRUG), dim3(64), 0, stream>>>(drug_e, lin_w, lin_b, y);

  gnn_bn<<<dim3(D), dim3(256), 0, stream>>>(y, bn_w, bn_b, (float*)d_out);
}